// VQLoss_82781199663436
// MI455X (gfx1250) — compile-verified
//
#include <hip/hip_runtime.h>

// Problem constants (from reference): B=8, Q=64, N=8192, K=512, C=256, GAMMA=0.25
#define PB 8
#define PQ 64
#define PN 8192
#define PK 512
#define PC 256

#define LSTR 68                 // padded LDS row stride (floats): conflict-free b64 reads
#define LDS_EMB_FLOATS (PK * LSTR)          // 34816 floats = 136 KB
#define LDS_TOTAL_FLOATS (LDS_EMB_FLOATS + PK)  // + emb2 (512 floats)

typedef __attribute__((ext_vector_type(2))) float v2f;
typedef __attribute__((ext_vector_type(8))) float v8f;

// ---------------------------------------------------------------------------
// Kernel 0: emb2[k] = sum_q emb[k][q]^2   (K=512 values)
// ---------------------------------------------------------------------------
__global__ __launch_bounds__(256) void emb2_kernel(const float* __restrict__ emb,
                                                   float* __restrict__ e2) {
    int k = blockIdx.x * blockDim.x + threadIdx.x;
    if (k < PK) {
        const float4* p = (const float4*)(emb + (size_t)k * PQ);
        float s = 0.f;
        #pragma unroll
        for (int i = 0; i < PQ / 4; ++i) {
            float4 v = p[i];
            s += v.x * v.x + v.y * v.y + v.z * v.z + v.w * v.w;
        }
        e2[k] = s;
    }
}

// ---------------------------------------------------------------------------
// Kernel 1: fused GEMM (emb x ze) + min-over-K + ze^2, via V_WMMA_F32_16X16X4_F32.
// emb (128KB) staged once per block into LDS (row stride 68 floats -> the
// per-wave A-fragment ds_load_b64 hits all 64 banks conflict-free).
// One wave owns a 16-wide n tile; 8 waves/block -> 128 n per block.
// Grid = B * (N/128) = 512 blocks. Dynamic LDS = 141312 bytes.
// ---------------------------------------------------------------------------
__global__ __launch_bounds__(256) void vq_min_kernel(const float* __restrict__ ze,
                                                     const float* __restrict__ emb,
                                                     const float* __restrict__ emb2,
                                                     float* __restrict__ partials) {
    extern __shared__ float lds[];            // [0, PK*LSTR): emb, then PK emb2
    float* lds_e2 = lds + LDS_EMB_FLOATS;

    const int lane = threadIdx.x & 31;
    const int wave = threadIdx.x >> 5;
    const int tilesPerB = PN / 128;           // 64
    const int b  = blockIdx.x / tilesPerB;
    const int tg = blockIdx.x % tilesPerB;
    const int n0 = tg * 128 + wave * 16;
    const int ncol = lane & 15;
    const int hi = lane >> 4;                 // 0: lanes 0-15, 1: lanes 16-31

    // ---- Stage emb into LDS, coalesced float4 reads, padded row-major store
    {
        const float4* src = (const float4*)emb;          // 8192 float4s
        for (int i = threadIdx.x; i < (PK * PQ) / 4; i += 256) {
            float4 v = src[i];
            int row = i >> 4;                 // / (PQ/4)
            int col = (i & 15) << 2;
            *(float4*)&lds[row * LSTR + col] = v;        // 16B-aligned ds_store_b128
        }
        for (int i = threadIdx.x; i < PK; i += 256) lds_e2[i] = emb2[i];
    }

    // ---- Preload B fragments (ze slice, 64 q x 16 n) + accumulate sum of squares
    const float* zb = ze + (size_t)b * PQ * PN + n0 + ncol;
    v2f bfrag[16];
    float sq = 0.f;
    #pragma unroll
    for (int s = 0; s < 16; ++s) {
        const int q0 = 4 * s + 2 * hi;
        float x = zb[q0 * PN];
        float y = zb[(q0 + 1) * PN];
        bfrag[s].x = x;
        bfrag[s].y = y;
        sq += x * x + y * y;                  // partial ||z||^2 (q-subset)
    }
    __syncthreads();

    // ---- Sweep K tiles: dual-accumulator WMMA, fuse min(emb2 - 2*dot)
    float minv = 3.402823466e38f;
    #pragma unroll 1
    for (int kt = 0; kt < PK / 16; ++kt) {
        const float* ea = &lds[(kt * 16 + ncol) * LSTR + 2 * hi];
        v8f c0 = {0.f, 0.f, 0.f, 0.f, 0.f, 0.f, 0.f, 0.f};
        v8f c1 = {0.f, 0.f, 0.f, 0.f, 0.f, 0.f, 0.f, 0.f};
        #pragma unroll
        for (int s = 0; s < 16; s += 2) {
            v2f a0, a1;
            a0.x = ea[4 * s];     a0.y = ea[4 * s + 1];
            a1.x = ea[4 * s + 4]; a1.y = ea[4 * s + 5];
            c0 = __builtin_amdgcn_wmma_f32_16x16x4_f32(
                     false, a0, false, bfrag[s],     (short)0, c0, false, false);
            c1 = __builtin_amdgcn_wmma_f32_16x16x4_f32(
                     false, a1, false, bfrag[s + 1], (short)0, c1, false, false);
        }
        const int krow = kt * 16 + 8 * hi;    // M rows this lane half sees
        #pragma unroll
        for (int r = 0; r < 8; ++r) {
            float cand = lds_e2[krow + r] - 2.0f * (c0[r] + c1[r]);
            minv = fminf(minv, cand);
        }
    }

    // ---- Merge lane halves (L and L+16 share the same n)
    float minOther = __shfl_xor(minv, 16, 32);
    float sqOther  = __shfl_xor(sq,  16, 32);
    // (1 + GAMMA) * min_k ||z - e_k||^2 ; value duplicated on both lanes of pair
    float total = 1.25f * ((sq + sqOther) + fminf(minv, minOther));

    // ---- Wave reduce (values duplicated x2 across lane pairs -> halve)
    #pragma unroll
    for (int off = 1; off < 32; off <<= 1) total += __shfl_xor(total, off, 32);
    total *= 0.5f;

    __shared__ float wsum[8];
    if (lane == 0) wsum[wave] = total;
    __syncthreads();
    if (threadIdx.x == 0) {
        float t = 0.f;
        #pragma unroll
        for (int i = 0; i < 8; ++i) t += wsum[i];
        partials[blockIdx.x] = t;
    }
}

// ---------------------------------------------------------------------------
// Kernel 2: reconstruction CE = logsumexp_c(qp[b,c,n]) - qp[b,target,n]
// Two-way-split streaming logsumexp (breaks serial fmax/exp/fma chain);
// thread per (b,n); fully coalesced over n. Grid = B*N/256 = 256 blocks.
// ---------------------------------------------------------------------------
__global__ __launch_bounds__(256) void ce_kernel(const float* __restrict__ qp,
                                                 const long long* __restrict__ tgt,
                                                 float* __restrict__ partials) {
    const int gid = blockIdx.x * blockDim.x + threadIdx.x;   // 0 .. B*N-1
    const int b = gid >> 13;                                  // / 8192
    const int n = gid & (PN - 1);
    const float* p = qp + (size_t)b * PC * PN + n;

    float m0 = -3.402823466e38f, s0 = 0.f;
    float m1 = -3.402823466e38f, s1 = 0.f;
    #pragma unroll 4
    for (int c = 0; c < PC / 2; ++c) {
        float x0 = p[c * PN];
        float x1 = p[(c + PC / 2) * PN];
        float mn0 = fmaxf(m0, x0);
        s0 = s0 * __expf(m0 - mn0) + __expf(x0 - mn0);
        m0 = mn0;
        float mn1 = fmaxf(m1, x1);
        s1 = s1 * __expf(m1 - mn1) + __expf(x1 - mn1);
        m1 = mn1;
    }
    float m = fmaxf(m0, m1);
    float s = s0 * __expf(m0 - m) + s1 * __expf(m1 - m);

    int t = (int)tgt[(size_t)b * PN + n];
    float rec = (m + __logf(s)) - p[(size_t)t * PN];

    #pragma unroll
    for (int off = 1; off < 32; off <<= 1) rec += __shfl_xor(rec, off, 32);
    __shared__ float wsum[8];
    if ((threadIdx.x & 31) == 0) wsum[threadIdx.x >> 5] = rec;
    __syncthreads();
    if (threadIdx.x == 0) {
        float t2 = 0.f;
        #pragma unroll
        for (int i = 0; i < 8; ++i) t2 += wsum[i];
        partials[blockIdx.x] = t2;
    }
}

// ---------------------------------------------------------------------------
// Kernel 3: deterministic final reduction of all partials -> d_out[0]
// ---------------------------------------------------------------------------
__global__ __launch_bounds__(256) void final_kernel(const float* __restrict__ partials,
                                                    int n, float* __restrict__ out) {
    float s = 0.f;
    for (int i = threadIdx.x; i < n; i += 256) s += partials[i];
    #pragma unroll
    for (int off = 1; off < 32; off <<= 1) s += __shfl_xor(s, off, 32);
    __shared__ float wsum[8];
    if ((threadIdx.x & 31) == 0) wsum[threadIdx.x >> 5] = s;
    __syncthreads();
    if (threadIdx.x == 0) {
        float t = 0.f;
        #pragma unroll
        for (int i = 0; i < 8; ++i) t += wsum[i];
        out[0] = t;
    }
}

// ---------------------------------------------------------------------------
extern "C" void kernel_launch(void* const* d_in, const int* in_sizes, int n_in,
                              void* d_out, int out_size, void* d_ws, size_t ws_size,
                              hipStream_t stream) {
    const float*     ze  = (const float*)d_in[0];      // (B,Q,N) f32
    const float*     emb = (const float*)d_in[1];      // (K,Q)   f32
    const float*     qp  = (const float*)d_in[2];      // (B,C,N) f32
    const long long* tw  = (const long long*)d_in[3];  // (B,N)   i64
    float* out = (float*)d_out;
    float* ws  = (float*)d_ws;

    float* e2 = ws;            // [0, 512)   : emb row norms
    float* p1 = ws + 512;      // [512,1024) : VQ partials (512 blocks)
    float* p2 = ws + 1024;     // [1024,1280): CE partials (256 blocks)

    const size_t vq_lds_bytes = (size_t)LDS_TOTAL_FLOATS * sizeof(float);  // 141312

    emb2_kernel  <<<2,   256, 0, stream>>>(emb, e2);
    vq_min_kernel<<<512, 256, vq_lds_bytes, stream>>>(ze, emb, e2, p1);
    ce_kernel    <<<256, 256, 0, stream>>>(qp, tw, p2);
    final_kernel <<<1,   256, 0, stream>>>(p1, 512 + 256, out);
}